// MultiGCN_11570641895563
// MI455X (gfx1250) — compile-verified
//
#include <hip/hip_runtime.h>
#include <hip/hip_bf16.h>

typedef __attribute__((ext_vector_type(2))) float v2f;
typedef __attribute__((ext_vector_type(8))) float v8f;

#define NN 50000
#define EE 800000
#define DD 128
#define GG 64
#define BSTRIDE 136   // LDS row stride (floats): 2*136 % 64 == 16 -> half-waves hit disjoint banks

__device__ __forceinline__ void atomAdd(float* p, float v) {
    __hip_atomic_fetch_add(p, v, __ATOMIC_RELAXED, __HIP_MEMORY_SCOPE_AGENT);
}

// ---------------- WMMA f32 GEMM: C[M,128] = A[M,K] @ B[K,128] (+bias, +relu) -------
// One block = 16-row strip; 8 waves = 8 column tiles; B staged in LDS (padded stride);
// K stepped by 4 with V_WMMA_F32_16X16X4_F32.
__global__ void k_gemm_wmma(const float* __restrict__ A, const float* __restrict__ B,
                            float* __restrict__ C, int K,
                            const float* __restrict__ bias, int relu) {
    extern __shared__ float Bs[];              // [K][BSTRIDE]
    const int tid = threadIdx.x;

    // cooperative load: B[K,128] -> LDS with padded row stride
    const int total4 = (K * DD) >> 2;
    for (int i = tid; i < total4; i += 256) {
        int base = i << 2;
        int kk = base >> 7;                    // row (DD == 128)
        int nn = base & 127;                   // col
        *(float4*)&Bs[kk * BSTRIDE + nn] = *(const float4*)&B[base];
    }
    __syncthreads();

    const int lane = tid & 31;
    const int tn   = tid >> 5;                 // column tile 0..7
    const int tm   = blockIdx.x;               // row tile
    const int half = lane >> 4;                // 0: K=k,k+1   1: K=k+2,k+3
    const int r    = lane & 15;
    const float* aPtr = A + (size_t)(tm * 16 + r) * K + 2 * half;
    const int bcol = tn * 16 + r;

    v8f acc = {};
    #pragma unroll 4
    for (int k = 0; k < K; k += 4) {
        v2f a = *(const v2f*)(aPtr + k);       // A[row][k+2*half .. +1]
        int kb = k + 2 * half;
        v2f b;
        b.x = Bs[kb * BSTRIDE + bcol];         // B[kb][col]
        b.y = Bs[(kb + 1) * BSTRIDE + bcol];   // B[kb+1][col]
        acc = __builtin_amdgcn_wmma_f32_16x16x4_f32(
            /*neg_a=*/false, a, /*neg_b=*/false, b,
            /*c_mod=*/(short)0, acc, /*reuse_a=*/false, /*reuse_b=*/false);
    }

    // D layout: VGPR j -> row (half*8 + j), col = tile_col + r
    const int row0 = tm * 16 + half * 8;
    const int col  = tn * 16 + r;
    #pragma unroll
    for (int j = 0; j < 8; ++j) {
        float v = acc[j];
        if (bias) {
            v += bias[col];
            if (relu) v = v > 0.f ? v : 0.f;
        }
        C[(size_t)(row0 + j) * DD + col] = v;
    }
}

// ---------------- elementwise / scatter kernels ----------------
__global__ void k_fill(float* __restrict__ p, float v, int n) {
    int i = blockIdx.x * blockDim.x + threadIdx.x;
    if (i < n) p[i] = v;
}

__global__ void k_deg_edges(const int* __restrict__ dst, float* __restrict__ deg, int E) {
    int e = blockIdx.x * blockDim.x + threadIdx.x;
    if (e < E) atomAdd(&deg[dst[e]], 1.0f);
}

__global__ void k_rsqrt(const float* __restrict__ deg, float* __restrict__ dinv, int n) {
    int i = blockIdx.x * blockDim.x + threadIdx.x;
    if (i < n) dinv[i] = rsqrtf(deg[i]);
}

// out[i][:] = h[i][:] * dinv[i]^2 + bias[:]   (self-loop term + bias)
__global__ void k_init_out(const float* __restrict__ h, const float* __restrict__ dinv,
                           const float* __restrict__ bias, float* __restrict__ out, int n) {
    long long tid = (long long)blockIdx.x * blockDim.x + threadIdx.x;
    int i = (int)(tid >> 5);
    if (i >= n) return;
    int d = ((int)(tid & 31)) * 4;
    float s = dinv[i];
    float inv = s * s;
    const float4 hv = *(const float4*)(h + (size_t)i * DD + d);
    float4 o;
    o.x = hv.x * inv + bias[d + 0];
    o.y = hv.y * inv + bias[d + 1];
    o.z = hv.z * inv + bias[d + 2];
    o.w = hv.w * inv + bias[d + 3];
    *(float4*)(out + (size_t)i * DD + d) = o;
}

// out[dst] += h[src] * dinv[src]*dinv[dst]   (32 lanes per edge, 4 floats per lane)
__global__ void k_edge_agg(const int* __restrict__ src, const int* __restrict__ dst,
                           const float* __restrict__ dinv, const float* __restrict__ h,
                           float* __restrict__ out, int E) {
    long long tid = (long long)blockIdx.x * blockDim.x + threadIdx.x;
    int e = (int)(tid >> 5);
    if (e >= E) return;
    int d = ((int)(tid & 31)) * 4;
    int s = src[e], t = dst[e];
    float norm = dinv[s] * dinv[t];
    const float4 hv = *(const float4*)(h + (size_t)s * DD + d);
    float* o = out + (size_t)t * DD + d;
    atomAdd(o + 0, hv.x * norm);
    atomAdd(o + 1, hv.y * norm);
    atomAdd(o + 2, hv.z * norm);
    atomAdd(o + 3, hv.w * norm);
}

__global__ void k_leaky(float* __restrict__ x, long long n) {
    long long i = (long long)blockIdx.x * blockDim.x + threadIdx.x;
    if (i < n) {
        float v = x[i];
        x[i] = v > 0.f ? v : 0.01f * v;
    }
}

__global__ void k_cnt(const int* __restrict__ batch, float* __restrict__ cnt, int n) {
    int i = blockIdx.x * blockDim.x + threadIdx.x;
    if (i < n) atomAdd(&cnt[batch[i]], 1.0f);
}

// pooled[batch[i]][colOfs + d] += x[i][d]   (pooled row stride = 256 for concat)
__global__ void k_pool_add(const float* __restrict__ x, const int* __restrict__ batch,
                           float* __restrict__ pooled, int n, int colOfs) {
    long long tid = (long long)blockIdx.x * blockDim.x + threadIdx.x;
    int i = (int)(tid >> 5);
    if (i >= n) return;
    int d = ((int)(tid & 31)) * 4;
    int g = batch[i];
    const float4 hv = *(const float4*)(x + (size_t)i * DD + d);
    float* o = pooled + (size_t)g * 256 + colOfs + d;
    atomAdd(o + 0, hv.x);
    atomAdd(o + 1, hv.y);
    atomAdd(o + 2, hv.z);
    atomAdd(o + 3, hv.w);
}

__global__ void k_pool_div(float* __restrict__ pooled, const float* __restrict__ cnt, int colOfs) {
    int idx = blockIdx.x * blockDim.x + threadIdx.x;
    if (idx >= GG * DD) return;
    int g = idx >> 7;
    int d = idx & 127;
    pooled[(size_t)g * 256 + colOfs + d] /= fmaxf(cnt[g], 1.0f);
}

// out[g] = dot(mlph[g,:], w2[:,0]) + b2[0]
__global__ void k_final(const float* __restrict__ mlph, const float* __restrict__ w2,
                        const float* __restrict__ b2, float* __restrict__ out) {
    int g = blockIdx.x * blockDim.x + threadIdx.x;
    if (g >= GG) return;
    float s = 0.f;
    #pragma unroll 8
    for (int j = 0; j < DD; ++j) s += mlph[(size_t)g * DD + j] * w2[j];
    out[g] = s + b2[0];
}

// ---------------- host orchestration ----------------
static void run_branch(const float* x, const int* src, const int* dst, const int* batch,
                       const float* W0, const float* b0, const float* W1, const float* b1,
                       float* buf0, float* buf1, float* deg, float* dinv,
                       float* pooled, float* cnt, int colOfs, hipStream_t stream) {
    const int TPB = 256;
    const int gN     = (NN + TPB - 1) / TPB;
    const int gE     = (EE + TPB - 1) / TPB;
    const long long nd = (long long)NN * DD;
    const int gND    = (int)((nd + TPB - 1) / TPB);
    const int gN32   = (int)(((long long)NN * 32 + TPB - 1) / TPB);
    const int gE32   = (int)(((long long)EE * 32 + TPB - 1) / TPB);
    const int gGemm  = NN / 16;                            // 3125 row strips
    const size_t lds = (size_t)DD * BSTRIDE * sizeof(float); // K=128 -> 69.6 KB

    // degree + symmetric norm (shared by both layers of this branch)
    k_fill<<<gN, TPB, 0, stream>>>(deg, 1.0f, NN);
    k_deg_edges<<<gE, TPB, 0, stream>>>(dst, deg, EE);
    k_rsqrt<<<gN, TPB, 0, stream>>>(deg, dinv, NN);

    // layer 0: x -> buf0 (gemm) -> buf1 (aggregate)
    k_gemm_wmma<<<gGemm, TPB, lds, stream>>>(x, W0, buf0, DD, nullptr, 0);
    k_init_out<<<gN32, TPB, 0, stream>>>(buf0, dinv, b0, buf1, NN);
    k_edge_agg<<<gE32, TPB, 0, stream>>>(src, dst, dinv, buf0, buf1, EE);
    k_leaky<<<gND, TPB, 0, stream>>>(buf1, nd);

    // layer 1: buf1 -> buf0 (gemm) -> buf1 (aggregate, reusing buf1 after gemm)
    k_gemm_wmma<<<gGemm, TPB, lds, stream>>>(buf1, W1, buf0, DD, nullptr, 0);
    k_init_out<<<gN32, TPB, 0, stream>>>(buf0, dinv, b1, buf1, NN);
    k_edge_agg<<<gE32, TPB, 0, stream>>>(src, dst, dinv, buf0, buf1, EE);
    k_leaky<<<gND, TPB, 0, stream>>>(buf1, nd);

    // mean pool into concat buffer columns [colOfs, colOfs+128)
    k_fill<<<1, GG, 0, stream>>>(cnt, 0.0f, GG);
    k_cnt<<<gN, TPB, 0, stream>>>(batch, cnt, NN);
    k_pool_add<<<gN32, TPB, 0, stream>>>(buf1, batch, pooled, NN, colOfs);
    k_pool_div<<<(GG * DD + TPB - 1) / TPB, TPB, 0, stream>>>(pooled, cnt, colOfs);
}

extern "C" void kernel_launch(void* const* d_in, const int* in_sizes, int n_in,
                              void* d_out, int out_size, void* d_ws, size_t ws_size,
                              hipStream_t stream) {
    const float* x1    = (const float*)d_in[0];
    const int*   ei1   = (const int*)d_in[1];
    const int*   bat1  = (const int*)d_in[2];
    const float* x2    = (const float*)d_in[3];
    const int*   ei2   = (const int*)d_in[4];
    const int*   bat2  = (const int*)d_in[5];
    const float* W1_0  = (const float*)d_in[6];
    const float* b1_0  = (const float*)d_in[7];
    const float* W1_1  = (const float*)d_in[8];
    const float* b1_1  = (const float*)d_in[9];
    const float* W2_0  = (const float*)d_in[10];
    const float* b2_0  = (const float*)d_in[11];
    const float* W2_1  = (const float*)d_in[12];
    const float* b2_1  = (const float*)d_in[13];
    const float* lin1W = (const float*)d_in[14];
    const float* lin1b = (const float*)d_in[15];
    const float* lin2W = (const float*)d_in[16];
    const float* lin2b = (const float*)d_in[17];

    float* ws     = (float*)d_ws;
    float* buf0   = ws;                               // [N,128]
    float* buf1   = buf0 + (size_t)NN * DD;           // [N,128]
    float* deg    = buf1 + (size_t)NN * DD;           // [N]
    float* dinv   = deg + NN;                         // [N]
    float* pooled = dinv + NN;                        // [64,256] concat(h1,h2)
    float* cnt    = pooled + (size_t)GG * 256;        // [64]
    float* mlph   = cnt + GG;                         // [64,128]

    const int TPB = 256;
    k_fill<<<(GG * 256 + TPB - 1) / TPB, TPB, 0, stream>>>(pooled, 0.0f, GG * 256);

    run_branch(x1, ei1, ei1 + EE, bat1, W1_0, b1_0, W1_1, b1_1,
               buf0, buf1, deg, dinv, pooled, cnt, 0, stream);
    run_branch(x2, ei2, ei2 + EE, bat2, W2_0, b2_0, W2_1, b2_1,
               buf0, buf1, deg, dinv, pooled, cnt, DD, stream);

    // lin1: [64,256] @ [256,128] + bias, ReLU  (WMMA, K=256 -> 139 KB LDS, OK on 320 KB WGP)
    const size_t lds1 = (size_t)(2 * DD) * BSTRIDE * sizeof(float);
    k_gemm_wmma<<<GG / 16, TPB, lds1, stream>>>(pooled, lin1W, mlph, 2 * DD, lin1b, 1);
    // lin2: [64,128] @ [128,1] + bias
    k_final<<<1, GG, 0, stream>>>(mlph, lin2W, lin2b, (float*)d_out);
}